// DifferentiableModalPlate_88948772700729
// MI455X (gfx1250) — compile-verified
//
#include <hip/hip_runtime.h>
#include <math.h>

typedef __attribute__((ext_vector_type(2))) float v2f;
typedef __attribute__((ext_vector_type(8))) float v8f;

#define KDT      (1.0f/44100.0f)
#define SRF      44100.0f
#define NMODES   6400
#define MGRID    80
#define TTILE    256
#define NCHUNK   20
#define CHUNK    (NMODES/NCHUNK)   // 320 modes per block
#define WAVES    8
#define PER_WAVE (CHUNK/WAVES)     // 40 modes per wave (multiple of 4)

__device__ __forceinline__ float softplus_f(float x) {
    return (x > 20.f) ? x : log1pf(__expf(x));
}
__device__ __forceinline__ float sigmoid_f(float x) {
    return 1.f / (1.f + __expf(-x));
}
// w*t mod 2pi, reduced in double (branchless; keeps all sin/cos args small so the
// hardware v_sin_f32/v_cos_f32 path is used -- no libm Payne-Hanek slow path).
__device__ __forceinline__ float red2pi(float w, float t) {
    double r = (double)w * (double)t;
    double q = floor(r * 0.15915494309189535);   // 1/(2pi)
    return (float)(r - q * 6.283185307179586);
}

// ---------------- Kernel 1: per-mode parameters + transposed B table (z^{16j}) ----------------
__global__ __launch_bounds__(256) void plate_setup(
    const float* mu_raw, const float* Dmu_raw, const float* T0mu_raw,
    const float* Ly_raw, const float* xo_raw, const float* yo_raw,
    float* __restrict__ modeW, float* __restrict__ modeD, float* __restrict__ modeC,
    float* __restrict__ BreT, float* __restrict__ BimT)   // [j][mode]
{
    int m = blockIdx.x * blockDim.x + threadIdx.x;
    if (m >= NMODES) return;
    const float PI = 3.14159265358979323846f;

    // scalar parameter transforms (redundant per thread, trivial cost)
    float mu    = softplus_f(mu_raw[0])   + 1e-4f;
    float Dmu   = softplus_f(Dmu_raw[0])  + 1e-4f;
    float T0mu  = softplus_f(T0mu_raw[0]) + 1e-4f;
    float Ly    = 1.1f + 2.9f * (tanhf(Ly_raw[0]) + 1.f) * 0.5f;
    float xo    = 0.49f + 0.51f * (tanhf(xo_raw[0]) + 1.f) * 0.5f;          // LX = 1
    float yo    = 0.51f * Ly + 0.49f * Ly * (tanhf(yo_raw[0]) + 1.f) * 0.5f;
    float xi = 0.335f, yi = 0.467f * Ly;

    float mv = (float)(m / MGRID + 1);
    float nv = (float)(m % MGRID + 1);

    float g1 = (mv * PI) * (mv * PI) + (nv * PI / Ly) * (nv * PI / Ly);
    float osq = T0mu * g1 + Dmu * g1 * g1;
    float omega = sqrtf(fmaxf(osq, 0.f));

    const float MAX_OM = 10000.f * 2.f * PI;
    float valid = sigmoid_f((MAX_OM - omega) * 0.01f) * sigmoid_f((omega - 40.f * PI) * 0.01f);

    float in_w  = __cosf(xi * PI * mv) * __cosf(yi * PI * nv / Ly);
    float out_w = __cosf(xo * PI * mv) * __cosf(yo * PI * nv / Ly);

    const float ALPHA = 1.15129254649702f;   // 3*ln10/tau0
    const float BETA  = 2.33261157e-7f;      // 3*ln10/(2pi*500)^2 * (1/2 - 1/6)
    float sigma = ALPHA + BETA * omega * omega;
    float ms = 0.25f * mu * Ly;
    float P = out_w * in_w * (KDT * KDT) * __expf(-sigma * KDT) / ms * valid;
    float coef = P / (__sinf(omega * KDT) + 1e-8f);

    float w = omega * KDT;   // rotation per sample
    float d = sigma * KDT;   // decay per sample
    modeW[m] = w; modeD[m] = d; modeC[m] = coef;

    // B[j][m] = z^{16 j}, time-tile invariant; stored transposed so the synth
    // kernel loads each lane's adjacent mode pair as one global_load_b64.
    for (int j = 0; j < 16; ++j) {
        float mag = __expf(-d * (16.f * j));
        float ang = red2pi(w, 16.f * (float)j);
        BreT[j * NMODES + m] = mag * __cosf(ang);
        BimT[j * NMODES + m] = mag * __sinf(ang);
    }
}

// ---------------- Kernel 0: zero the disp accumulator ----------------
__global__ void plate_zero(float* __restrict__ disp, int n) {
    int t = blockIdx.x * blockDim.x + threadIdx.x;
    if (t < n) disp[t] = 0.f;
}

// ---------------- Kernel 2: complex-GEMM synthesis via WMMA f32 16x16x4 ----------------
__global__ __launch_bounds__(256) void plate_synth(
    const float* __restrict__ modeW, const float* __restrict__ modeD, const float* __restrict__ modeC,
    const float* __restrict__ BreT, const float* __restrict__ BimT,
    float* __restrict__ disp, int T)
{
    __shared__ float4 sPar[CHUNK];          // {w, d, coef, phase0}
    __shared__ float  red[WAVES][TTILE];

    int tid = threadIdx.x;
    int tile = blockIdx.x;                  // 256-sample time tile
    int chunkBase = blockIdx.y * CHUNK;
    float t0 = (float)(tile * TTILE);

    // Stage per-mode params; fold w*t0 mod 2pi (double, branchless) so all
    // inner-loop trig args are small.
    for (int i = tid; i < CHUNK; i += 256) {
        float w = modeW[chunkBase + i];
        float d = modeD[chunkBase + i];
        float c = modeC[chunkBase + i];
        sPar[i] = make_float4(w, d, c, red2pi(w, t0));
    }
    __syncthreads();

    int wave = tid >> 5, lane = tid & 31;
    int half = lane >> 4, li = lane & 15;
    float trow = t0 + (float)li;            // A row time t0+i
    float lip1 = (float)(li + 1);

    const v2f* BreV = (const v2f*)BreT;
    const v2f* BimV = (const v2f*)BimT;

    v8f acc = {};
    int mbase = wave * PER_WAVE;
    for (int s = 0; s < PER_WAVE; s += 4) {
        // A layout (16x4 f32): lanes 0-15 hold K=0,1; lanes 16-31 hold K=2,3; row M = lane&15
        int k0 = mbase + s + 2 * half;      // even by construction
        float4 p0 = sPar[k0];
        float4 p1 = sPar[k0 + 1];

        float e0 = p0.z * __expf(-p0.y * trow);
        float e1 = p1.z * __expf(-p1.y * trow);
        float a0 = p0.w + p0.x * lip1;      // = w*(t0+li+1) mod 2pi (+ small)
        float a1 = p1.w + p1.x * lip1;
        v2f Are = { e0 * __cosf(a0), e1 * __cosf(a1) };
        v2f Aim = { e0 * __sinf(a0), e1 * __sinf(a1) };

        // B layout (4x16 f32): lane holds column N = li, K split as in A.
        // BreT[j][mode]: modes k0,k0+1 adjacent -> single b64 load each.
        int bidx = li * (NMODES / 2) + ((chunkBase + k0) >> 1);
        v2f Br = BreV[bidx];
        v2f Bi = BimV[bidx];

        // Im(A x B) = Are*Bim + Aim*Bre : two f32 WMMA accumulations
        acc = __builtin_amdgcn_wmma_f32_16x16x4_f32(false, Are, false, Bi, (short)0, acc, false, false);
        acc = __builtin_amdgcn_wmma_f32_16x16x4_f32(false, Aim, false, Br, (short)0, acc, false, false);
    }

    // C/D layout: M = v + 8*half, N = li ; t_local = M + 16*N
#pragma unroll
    for (int v = 0; v < 8; ++v) {
        int rowi = v + 8 * half;
        red[wave][rowi + 16 * li] = acc[v];
    }
    __syncthreads();

    float sum = 0.f;
#pragma unroll
    for (int w2 = 0; w2 < WAVES; ++w2) sum += red[w2][tid];
    int t = tile * TTILE + tid;
    if (t < T) atomicAdd(&disp[t], sum);
}

// ---------------- Kernel 3: fused first-difference + peak normalize ----------------
__global__ __launch_bounds__(256) void plate_finalize(
    const float* __restrict__ disp, float* __restrict__ out, int T)
{
    __shared__ float smax[256];
    int tid = threadIdx.x;
    float lmax = 0.f;
    for (int t = tid; t < T; t += 256) {
        float prev = (t == 0) ? 0.f : disp[t - 1];
        float ir = (disp[t] - prev) * SRF;
        out[t] = ir;
        lmax = fmaxf(lmax, fabsf(ir));
    }
    smax[tid] = lmax;
    __syncthreads();
    for (int o = 128; o > 0; o >>= 1) {
        if (tid < o) smax[tid] = fmaxf(smax[tid], smax[tid + o]);
        __syncthreads();
    }
    float inv = 1.f / (smax[0] + 1e-8f);
    for (int t = tid; t < T; t += 256) out[t] *= inv;
}

extern "C" void kernel_launch(void* const* d_in, const int* in_sizes, int n_in,
                              void* d_out, int out_size, void* d_ws, size_t ws_size,
                              hipStream_t stream)
{
    const float* mu   = (const float*)d_in[0];
    const float* Dmu  = (const float*)d_in[1];
    const float* T0mu = (const float*)d_in[2];
    const float* Lyr  = (const float*)d_in[3];
    const float* xor_ = (const float*)d_in[4];
    const float* yor_ = (const float*)d_in[5];
    // d_in[6] = num_samples (int); out_size carries the same value host-side.

    float* ws    = (float*)d_ws;
    float* modeW = ws;
    float* modeD = modeW + NMODES;
    float* modeC = modeD + NMODES;
    float* BreT  = modeC + NMODES;
    float* BimT  = BreT + NMODES * 16;
    float* disp  = BimT + NMODES * 16;

    int T = out_size;
    int numTiles = (T + TTILE - 1) / TTILE;
    int dispN = numTiles * TTILE;

    plate_zero<<<(dispN + 255) / 256, 256, 0, stream>>>(disp, dispN);
    plate_setup<<<(NMODES + 255) / 256, 256, 0, stream>>>(mu, Dmu, T0mu, Lyr, xor_, yor_,
                                                          modeW, modeD, modeC, BreT, BimT);
    dim3 grid(numTiles, NCHUNK);
    plate_synth<<<grid, 256, 0, stream>>>(modeW, modeD, modeC, BreT, BimT, disp, T);
    plate_finalize<<<1, 256, 0, stream>>>(disp, (float*)d_out, T);
}